// GCN1_24257975288509
// MI455X (gfx1250) — compile-verified
//
#include <hip/hip_runtime.h>
#include <hip/hip_bf16.h>

#define N_NODES 50000
#define N_EDGES 800000
#define F_IN    128
#define H_DIM   304

typedef __attribute__((ext_vector_type(2))) float v2f;
typedef __attribute__((ext_vector_type(8))) float v8f;

// ---------------- utility kernels ----------------

__global__ void gcn_zero_f32(float* __restrict__ p, long long n) {
  long long i = (long long)blockIdx.x * blockDim.x + threadIdx.x;
  if (i < n) p[i] = 0.0f;
}

__global__ void gcn_degree_count(const int* __restrict__ src, const int* __restrict__ dst,
                                 float* __restrict__ degS, float* __restrict__ degD, int E) {
  int e = blockIdx.x * blockDim.x + threadIdx.x;
  if (e < E) {
    atomicAdd(&degS[src[e]], 1.0f);
    atomicAdd(&degD[dst[e]], 1.0f);
  }
}

// deg -> clamp(deg,1)^-0.5 ; operates on 2N contiguous floats (src-norm then dst-norm)
__global__ void gcn_deg_to_norm(float* __restrict__ p, int n) {
  int i = blockIdx.x * blockDim.x + threadIdx.x;
  if (i < n) {
    float d = p[i];
    d = d < 1.0f ? 1.0f : d;
    p[i] = 1.0f / sqrtf(d);
  }
}

// ---------------- SpMM: edge-parallel scatter-add with fused src-norm ----------------
// out[dst[e]][:] += X[src[e]][:] * dn_src[src[e]]
// One thread per (edge, 4-float chunk); consecutive lanes -> consecutive 16B chunks
// of the same source row (coalesced). F is a compile-time constant so the
// edge/chunk split is a mul-shift, not a runtime division. Atomics land mostly
// in the 192MB L2 since the 61MB destination matrix is L2-resident.
template <int F>
__global__ __launch_bounds__(256) void gcn_spmm_scatter(
    const float* __restrict__ X, const int* __restrict__ src,
    const int* __restrict__ dst, const float* __restrict__ dn_src,
    float* __restrict__ out, int E) {
  constexpr int chunks = F >> 2;
  unsigned int t = blockIdx.x * blockDim.x + threadIdx.x;   // E*chunks < 2^31
  unsigned int total = (unsigned int)E * chunks;
  if (t >= total) return;
  unsigned int e = t / chunks;           // constant divisor -> mul-shift
  unsigned int c = t - e * chunks;
  int s = src[e];
  int d = dst[e];
  float sc = dn_src[s];
  const float4 v = *((const float4*)(X + (long long)s * F) + c);
  float* op = out + (long long)d * F + (c << 2);
  atomicAdd(op + 0, v.x * sc);
  atomicAdd(op + 1, v.y * sc);
  atomicAdd(op + 2, v.z * sc);
  atomicAdd(op + 3, v.w * sc);
}

// ---------------- GEMM: out = relu( diag(dnD) * (A @ W) + b ), fp32 WMMA ----------------
// Each wave computes a 16x64 strip of the output (4 accumulator tiles, A reused).
// A: [M x K] row-major, W: [K x H] row-major, out: [M x H] row-major.
// A-fragment layout (V_WMMA_F32_16X16X4_F32): lane<16 holds K={0,1}, lane>=16 holds K={2,3}.
// All inner-loop addresses are maintained as incremented pointers (constant strides),
// and the dnorm_dst row scale is applied in the epilogue, off the load->wmma path.
__global__ __launch_bounds__(256) void gcn_gemm_wmma_f32(
    const float* __restrict__ A, const float* __restrict__ W,
    const float* __restrict__ bias, const float* __restrict__ dnD,
    float* __restrict__ out, int M, int K, int H) {
  const int nStrips = (H + 63) >> 6;
  int waveId = blockIdx.x * (blockDim.x >> 5) + (threadIdx.x >> 5);
  int lane = threadIdx.x & 31;
  int totalWaves = (M >> 4) * nStrips;
  if (waveId >= totalWaves) return;   // wave-uniform

  int mTile = waveId / nStrips;
  int strip = waveId - mTile * nStrips;

  int l16  = lane & 15;
  int hi   = lane >> 4;   // 0 or 1
  int koff = hi << 1;     // 0 or 2

  int aRow = (mTile << 4) + l16;
  const float* ap = A + (long long)aRow * K + koff;   // 8B-aligned (koff in {0,2})

  int colBase = (strip << 6) + l16;
  int  col[4];
  bool valid[4];
#pragma unroll
  for (int c = 0; c < 4; ++c) {
    int cc = colBase + (c << 4);
    valid[c] = cc < H;           // uniform per tile (H multiple of 16)
    col[c] = valid[c] ? cc : 0;  // clamped for safe loads
  }

  const float* bp0 = W + (long long)koff * H + col[0];
  const float* bp1 = W + (long long)koff * H + col[1];
  const float* bp2 = W + (long long)koff * H + col[2];
  const float* bp3 = W + (long long)koff * H + col[3];
  const long long bStep = 4LL * H;   // floats per K-step

  v8f acc0 = {}, acc1 = {}, acc2 = {}, acc3 = {};

  for (int k = 0; k < K; k += 4) {
    v2f a = *(const v2f*)ap;                    // A[aRow][k+koff .. k+koff+1]
    v2f b0, b1, b2, b3;
    b0.x = bp0[0];  b0.y = bp0[H];
    b1.x = bp1[0];  b1.y = bp1[H];
    b2.x = bp2[0];  b2.y = bp2[H];
    b3.x = bp3[0];  b3.y = bp3[H];
    acc0 = __builtin_amdgcn_wmma_f32_16x16x4_f32(false, a, false, b0, (short)0, acc0, false, false);
    acc1 = __builtin_amdgcn_wmma_f32_16x16x4_f32(false, a, false, b1, (short)0, acc1, false, false);
    acc2 = __builtin_amdgcn_wmma_f32_16x16x4_f32(false, a, false, b2, (short)0, acc2, false, false);
    acc3 = __builtin_amdgcn_wmma_f32_16x16x4_f32(false, a, false, b3, (short)0, acc3, false, false);
    ap  += 4;
    bp0 += bStep; bp1 += bStep; bp2 += bStep; bp3 += bStep;
  }

  // epilogue: row scale (dnorm_dst), bias, relu
  float rs[8];
#pragma unroll
  for (int r = 0; r < 8; ++r) rs[r] = dnD[(mTile << 4) + (hi << 3) + r];

  v8f accs[4] = {acc0, acc1, acc2, acc3};
#pragma unroll
  for (int c = 0; c < 4; ++c) {
    if (!valid[c]) continue;
    float bv = bias[col[c]];
#pragma unroll
    for (int r = 0; r < 8; ++r) {
      int orow = (mTile << 4) + (hi << 3) + r;
      float v = accs[c][r] * rs[r] + bv;
      out[(long long)orow * H + col[c]] = v > 0.0f ? v : 0.0f;
    }
  }
}

// ---------------- reduction ----------------

__global__ __launch_bounds__(256) void gcn_reduce_sum(const float* __restrict__ in,
                                                      long long n, float* __restrict__ acc) {
  __shared__ float sm[256];
  long long i = (long long)blockIdx.x * blockDim.x + threadIdx.x;
  long long stride = (long long)gridDim.x * blockDim.x;
  float s = 0.0f;
  for (; i < n; i += stride) s += in[i];
  sm[threadIdx.x] = s;
  __syncthreads();
  for (int off = 128; off > 0; off >>= 1) {
    if (threadIdx.x < off) sm[threadIdx.x] += sm[threadIdx.x + off];
    __syncthreads();
  }
  if (threadIdx.x == 0) atomicAdd(acc, sm[0]);
}

__global__ void gcn_finalize(const float* __restrict__ acc, float* __restrict__ out) {
  if (blockIdx.x == 0 && threadIdx.x == 0)
    out[0] = acc[0] * (1.0f / (4.0f * (float)N_NODES * (float)H_DIM));
}

// ---------------- launch ----------------

extern "C" void kernel_launch(void* const* d_in, const int* in_sizes, int n_in,
                              void* d_out, int out_size, void* d_ws, size_t ws_size,
                              hipStream_t stream) {
  const int N = N_NODES, E = N_EDGES, H = H_DIM;
  const float* Wm[3] = {(const float*)d_in[12], (const float*)d_in[14], (const float*)d_in[16]};
  const float* bm[3] = {(const float*)d_in[13], (const float*)d_in[15], (const float*)d_in[17]};

  float* ws   = (float*)d_ws;
  float* dnS  = ws;                       // N  (src-degree norm)
  float* dnD  = dnS + N;                  // N  (dst-degree norm)
  float* bufA = dnD + N;                  // N*H (SpMM accumulator)
  float* bufB = bufA + (size_t)N * H;     // N*H (layer output)
  float* acc  = bufB + (size_t)N * H;     // 1  (global sum)

  auto zero = [&](float* p, long long n) {
    int blocks = (int)((n + 255) / 256);
    gcn_zero_f32<<<blocks, 256, 0, stream>>>(p, n);
  };

  zero(acc, 1);

  const int gemmWaves  = (N / 16) * ((H + 63) / 64);
  const int gemmBlocks = (gemmWaves + 7) / 8;

  for (int g = 0; g < 4; ++g) {
    const float* X   = (const float*)d_in[3 * g + 0];
    const int*   src = (const int*)d_in[3 * g + 1];
    const int*   dst = (const int*)d_in[3 * g + 2];

    // degree norms for this graph (shared by all 3 layers)
    zero(dnS, 2LL * N);
    gcn_degree_count<<<(E + 255) / 256, 256, 0, stream>>>(src, dst, dnS, dnD, E);
    gcn_deg_to_norm<<<(2 * N + 255) / 256, 256, 0, stream>>>(dnS, 2 * N);

    // layer 1: K = F_IN
    zero(bufA, (long long)N * F_IN);
    {
      unsigned int tot = (unsigned int)E * (F_IN / 4);
      gcn_spmm_scatter<F_IN><<<(tot + 255) / 256, 256, 0, stream>>>(X, src, dst, dnS, bufA, E);
    }
    gcn_gemm_wmma_f32<<<gemmBlocks, 256, 0, stream>>>(bufA, Wm[0], bm[0], dnD, bufB, N, F_IN, H);

    // layers 2 and 3: K = H
    for (int l = 1; l < 3; ++l) {
      zero(bufA, (long long)N * H);
      unsigned int tot = (unsigned int)E * (H / 4);
      gcn_spmm_scatter<H_DIM><<<(tot + 255) / 256, 256, 0, stream>>>(bufB, src, dst, dnS, bufA, E);
      gcn_gemm_wmma_f32<<<gemmBlocks, 256, 0, stream>>>(bufA, Wm[l], bm[l], dnD, bufB, N, H, H);
    }

    // accumulate sum of this graph's final node features
    gcn_reduce_sum<<<2048, 256, 0, stream>>>(bufB, (long long)N * H, acc);
  }

  gcn_finalize<<<1, 64, 0, stream>>>(acc, (float*)d_out);
}